// QuantEmbedding_70935679860876
// MI455X (gfx1250) — compile-verified
//
#include <hip/hip_runtime.h>

// QIL quantized embedding gather for MI455X (gfx1250).
// Bandwidth-bound: ~101 MB traffic -> ~4.3 us at 23.3 TB/s. No matmul => no
// WMMA. Uses the CDNA5 async global->LDS copy path (ASYNCcnt): each block
// issues ALL 8 gathered-row copies up front (async loads retire in order),
// then drains with descending s_wait_asynccnt immediates, overlapping the
// quantize math + NT stores of row r with the HBM fetch of rows r+1..7.

#define DIM   768              // embedding dim (reference-fixed)
#define VEC   4                // float4 per thread
#define TPB   (DIM / VEC)      // 192 threads = 6 wave32
#define ROWS  8                // token rows per block, 8-deep async pipeline

typedef float v4f __attribute__((ext_vector_type(4)));

__global__ __launch_bounds__(TPB) void qil_embed_gather_kernel(
    const int* __restrict__ x,      // [n_idx] token ids
    const float* __restrict__ wgt,  // [vocab, DIM]
    const float* __restrict__ pp,   // [1] pruning point
    const float* __restrict__ cp,   // [1] clipping point
    float* __restrict__ out,        // [n_idx, DIM] ++ [s, p]
    int n_idx, long long tail_off)
{
    __shared__ __align__(16) float buf[ROWS][DIM];   // 24 KB, written once each

    const int tid  = threadIdx.x;
    const int row0 = blockIdx.x * ROWS;
    const int nrows = min(ROWS, n_idx - row0);
    if (nrows <= 0) return;

    // Uniform quantizer parameters (N = 2^(8-1)-1 = 127).
    const float p     = fmaxf(pp[0], 0.0f);
    const float c     = cp[0];
    const float s     = 127.0f / (c - p);
    const float inv_s = (c - p) * (1.0f / 127.0f);

    // Per-lane 16B async copy: gathered row chunk -> LDS slot. Generic
    // pointers into LDS carry the LDS byte offset in addr[31:0] (ISA 10.2),
    // which is exactly the per-lane VDST address the async op needs.
    auto issue = [&](int r, int idx) {
        const uint32_t lo = (uint32_t)(uintptr_t)(&buf[r][tid * VEC]);
        const uint64_t ga = (uint64_t)(uintptr_t)
            (wgt + (size_t)idx * DIM + (size_t)tid * VEC);
        asm volatile("global_load_async_to_lds_b128 %0, %1, off"
                     :: "v"(lo), "v"(ga) : "memory");
    };

    // Quantize one row from LDS and stream it out (each thread reads back
    // exactly the bytes its own lane copied: no barrier, just ASYNCcnt).
    auto drain = [&](int r) {
        const v4f wv = *(const v4f*)(&buf[r][tid * VEC]);
        v4f o;
#pragma unroll
        for (int k = 0; k < VEC; ++k) {
            const float w   = wv[k];
            const float aw  = fabsf(w);
            const float sgn = (w > 0.0f) ? 1.0f : ((w < 0.0f) ? -1.0f : 0.0f);
            float mag = (aw < p) ? 0.0f : (aw - p) * s;   // prune + affine
            mag = rintf(fminf(mag, 127.0f));               // clip + RTE round
            // sign(round(wq)): values that round to 0 must output exactly 0.
            o[k] = (mag == 0.0f) ? 0.0f : (mag * inv_s + p) * sgn;
        }
        // Single-touch streaming output: NT b128 store keeps the ~42 MB of
        // unique gathered rows resident in the 192 MB L2 instead.
        float* dst = out + (size_t)(row0 + r) * DIM + (size_t)tid * VEC;
        __builtin_nontemporal_store(o, (v4f*)dst);
    };

    if (nrows == ROWS) {
        // Hoist the 8 consecutive uniform index loads -> one wide s_load.
        int idx[ROWS];
#pragma unroll
        for (int r = 0; r < ROWS; ++r) idx[r] = x[row0 + r];

        // Fill: 8 async copies in flight (ASYNCcnt is 6 bits; retire in order).
#pragma unroll
        for (int r = 0; r < ROWS; ++r) issue(r, idx[r]);

        // Drain: row r ready once ASYNCcnt <= ROWS-1-r; rows r+1.. still fly.
#pragma unroll
        for (int r = 0; r < ROWS; ++r) {
            asm volatile("s_wait_asynccnt %0" :: "n"(ROWS - 1 - r) : "memory");
            drain(r);
        }
    } else {
        // Rare partial tail block (n_idx % ROWS != 0): conservative waits.
        for (int r = 0; r < nrows; ++r) {
            issue(r, x[row0 + r]);
            asm volatile("s_wait_asynccnt 0x0" ::: "memory");
            drain(r);
        }
    }

    if (blockIdx.x == 0 && tid == 0) {
        out[tail_off]     = s;   // weight_scaling_factor
        out[tail_off + 1] = p;   // clamped pruning point
    }
}

extern "C" void kernel_launch(void* const* d_in, const int* in_sizes, int n_in,
                              void* d_out, int out_size, void* d_ws, size_t ws_size,
                              hipStream_t stream) {
    const int*   x   = (const int*)d_in[0];     // token ids (int32)
    const float* wgt = (const float*)d_in[1];   // [vocab, 768] fp32
    const float* pp  = (const float*)d_in[2];   // pruning point [1]
    const float* cp  = (const float*)d_in[3];   // clipping point [1]
    float* out = (float*)d_out;

    const int n_idx = in_sizes[0];              // 8*2048 = 16384
    const long long tail = (long long)out_size - 2;

    const int grid = (n_idx + ROWS - 1) / ROWS; // 2048 blocks of 6 waves
    qil_embed_gather_kernel<<<grid, TPB, 0, stream>>>(x, wgt, pp, cp, out,
                                                      n_idx, tail);
}